// SepNet_90744069030474
// MI455X (gfx1250) — compile-verified
//
#include <hip/hip_runtime.h>
#include <hip/hip_bf16.h>
#include <math.h>

// ---- Problem constants (fixed by reference) ----
#define NN     50000      // nodes
#define FD     64         // features
#define HH_    4          // heads
#define EE     300000     // edges
#define ESN    400000     // scatter edges
#define GG     128        // graphs
#define EA     (EE + NN)  // edges + self loops = 350000
#define HF     256        // H*F
#define EPS    1e-5f
#define NSLOPE 0.2f

typedef __attribute__((ext_vector_type(2))) float v2f;
typedef __attribute__((ext_vector_type(8))) float v8f;

// ---------- helpers ----------
__device__ __forceinline__ unsigned fenc(float f) {
    unsigned u = __float_as_uint(f);
    return (u & 0x80000000u) ? ~u : (u | 0x80000000u);
}
__device__ __forceinline__ float fdec(unsigned u) {
    return __uint_as_float((u & 0x80000000u) ? (u ^ 0x80000000u) : ~u);
}

__global__ void fill_u32_kernel(unsigned* p, unsigned v, int n) {
    int i = blockIdx.x * blockDim.x + threadIdx.x;
    if (i < n) p[i] = v;
}

// ---------- solo pass: edge-weighted scatter add (16 threads / edge, float4) ----------
__global__ void solo_kernel(const int* __restrict__ s_idx, const int* __restrict__ t_idx,
                            const float* __restrict__ w, const float* __restrict__ h,
                            float* __restrict__ out, int nedges, int abs_in) {
    int tid = blockIdx.x * blockDim.x + threadIdx.x;
    int e = tid >> 4;
    if (e >= nedges) return;
    int c = (tid & 15) << 2;
    int s = s_idx[e], t = t_idx[e];
    float ww = w[e];
    float4 v = *(const float4*)(h + (size_t)s * FD + c);
    if (abs_in) { v.x = fabsf(v.x); v.y = fabsf(v.y); v.z = fabsf(v.z); v.w = fabsf(v.w); }
    float* op = out + (size_t)t * FD + c;
    atomicAdd(op + 0, ww * v.x);
    atomicAdd(op + 1, ww * v.y);
    atomicAdd(op + 2, ww * v.z);
    atomicAdd(op + 3, ww * v.w);
}

// ---------- GEMM1 (WMMA f32): HH = A(NN x 64) @ B(64 x 256) ----------
// one wave per 16x16 output tile; 3125 M-tiles * 16 N-tiles = 50000 waves (exact)
__global__ void gemm1_wmma(const float* __restrict__ A, const float* __restrict__ B,
                           float* __restrict__ D) {
    int wave = blockIdx.x * (blockDim.x >> 5) + (threadIdx.x >> 5);
    int lane = threadIdx.x & 31;
    int mt = wave >> 4;          // 0..3124
    int nt = wave & 15;          // 0..15
    int mrow = mt * 16 + (lane & 15);
    int ncol = nt * 16 + (lane & 15);
    int hi = lane >> 4;          // 0: K pair {0,1}, 1: K pair {2,3}
    v8f acc = {};
    for (int k0 = 0; k0 < FD; k0 += 4) {
        int ka = k0 + 2 * hi;
        v2f a, b;
        a.x = A[(size_t)mrow * FD + ka];
        a.y = A[(size_t)mrow * FD + ka + 1];
        b.x = B[(size_t)ka * HF + ncol];
        b.y = B[(size_t)(ka + 1) * HF + ncol];
        acc = __builtin_amdgcn_wmma_f32_16x16x4_f32(false, a, false, b,
                                                    (short)0, acc, false, false);
    }
#pragma unroll
    for (int v = 0; v < 8; v++) {
        int m = mt * 16 + v + 8 * hi;
        D[(size_t)m * HF + ncol] = acc[v];
    }
}

// ---------- per-node attention scores ----------
__global__ void scores_kernel(const float* __restrict__ HHp, const float* __restrict__ gas,
                              const float* __restrict__ gad, float* __restrict__ asrc,
                              float* __restrict__ adst) {
    int tid = blockIdx.x * blockDim.x + threadIdx.x;
    if (tid >= NN * HH_) return;
    int n = tid >> 2, h = tid & 3;
    const float* hp = HHp + (size_t)n * HF + h * FD;
    const float* sp = gas + h * FD;
    const float* dp = gad + h * FD;
    float s = 0.f, d = 0.f;
#pragma unroll 4
    for (int f = 0; f < FD; f++) { float v = hp[f]; s += v * sp[f]; d += v * dp[f]; }
    asrc[tid] = s;
    adst[tid] = d;
}

// ---------- edge pass 1: leakyrelu score + segment max (encoded u32 atomicMax) ----------
__global__ void edge_max_kernel(const int* __restrict__ ei, const float* __restrict__ asrc,
                                const float* __restrict__ adst, float* __restrict__ esc,
                                unsigned* __restrict__ maxv) {
    int tid = blockIdx.x * blockDim.x + threadIdx.x;
    if (tid >= EA * HH_) return;
    int j = tid >> 2, h = tid & 3;
    int s, d;
    if (j < EE) { s = ei[j]; d = ei[EE + j]; } else { s = j - EE; d = s; }
    float e = asrc[s * HH_ + h] + adst[d * HH_ + h];
    e = (e > 0.f) ? e : NSLOPE * e;
    esc[tid] = e;
    atomicMax(maxv + (size_t)d * HH_ + h, fenc(e));
}

// ---------- edge pass 2: exp(e - max), segment sum ----------
__global__ void edge_exp_kernel(const int* __restrict__ ei, float* __restrict__ esc,
                                const unsigned* __restrict__ maxv, float* __restrict__ sumv) {
    int tid = blockIdx.x * blockDim.x + threadIdx.x;
    if (tid >= EA * HH_) return;
    int j = tid >> 2, h = tid & 3;
    int d = (j < EE) ? ei[EE + j] : (j - EE);
    float m = fdec(maxv[(size_t)d * HH_ + h]);
    float ex = expf(esc[tid] - m);
    esc[tid] = ex;
    atomicAdd(sumv + (size_t)d * HH_ + h, ex);
}

// ---------- edge pass 3: scatter hh[src] * alpha into dst (64 threads / edge) ----------
__global__ void edge_scatter_kernel(const int* __restrict__ ei, const float* __restrict__ HHp,
                                    const float* __restrict__ esc, const float* __restrict__ sumv,
                                    float* __restrict__ outp) {
    long long tid = (long long)blockIdx.x * blockDim.x + threadIdx.x;
    int j = (int)(tid >> 6);
    if (j >= EA) return;
    int t = (int)(tid & 63);
    int h = t >> 4;
    int c = (t & 15) << 2;
    int s, d;
    if (j < EE) { s = ei[j]; d = ei[EE + j]; } else { s = j - EE; d = s; }
    float alpha = esc[(size_t)j * HH_ + h] / sumv[(size_t)d * HH_ + h];
    float4 v = *(const float4*)(HHp + (size_t)s * HF + h * FD + c);
    float* op = outp + (size_t)d * HF + h * FD + c;
    atomicAdd(op + 0, alpha * v.x);
    atomicAdd(op + 1, alpha * v.y);
    atomicAdd(op + 2, alpha * v.z);
    atomicAdd(op + 3, alpha * v.w);
}

// ---------- GEMM2 (WMMA f32): XS2 += BN( elu(ATT+gb)(NN x 256) @ mW(256 x 64) + mb ) ----------
// one wave per 16x16 tile; 3125 M-tiles * 4 N-tiles = 12500 waves; 4 waves/block (exact)
__global__ void gemm2_wmma(const float* __restrict__ ATT, const float* __restrict__ gb,
                           const float* __restrict__ Bw, const float* __restrict__ mb,
                           const float* __restrict__ mg, const float* __restrict__ mbe,
                           const float* __restrict__ mrm, const float* __restrict__ mrv,
                           float* __restrict__ XS2) {
    int wave = blockIdx.x * 4 + (threadIdx.x >> 5);
    int lane = threadIdx.x & 31;
    int mt = wave >> 2;          // 0..3124
    int nt = wave & 3;           // 0..3
    int mrow = mt * 16 + (lane & 15);
    int ncol = nt * 16 + (lane & 15);
    int hi = lane >> 4;
    v8f acc = {};
    for (int k0 = 0; k0 < HF; k0 += 4) {
        int ka = k0 + 2 * hi;
        float a0 = ATT[(size_t)mrow * HF + ka] + gb[ka];
        float a1 = ATT[(size_t)mrow * HF + ka + 1] + gb[ka + 1];
        v2f a, b;
        a.x = (a0 > 0.f) ? a0 : (expf(a0) - 1.0f);   // ELU
        a.y = (a1 > 0.f) ? a1 : (expf(a1) - 1.0f);
        b.x = Bw[(size_t)ka * FD + ncol];
        b.y = Bw[(size_t)(ka + 1) * FD + ncol];
        acc = __builtin_amdgcn_wmma_f32_16x16x4_f32(false, a, false, b,
                                                    (short)0, acc, false, false);
    }
    float g = mg[ncol], be = mbe[ncol], rm = mrm[ncol];
    float rs = rsqrtf(mrv[ncol] + EPS);
    float bb = mb[ncol];
#pragma unroll
    for (int v = 0; v < 8; v++) {
        int m = mt * 16 + v + 8 * hi;
        float h2 = acc[v] + bb;
        XS2[(size_t)m * FD + ncol] += g * (h2 - rm) * rs + be;
    }
}

// ---------- global add pool (16 threads / node, float4) ----------
__global__ void pool_kernel(const float* __restrict__ XS2, const int* __restrict__ batch,
                            float* __restrict__ pool) {
    int tid = blockIdx.x * blockDim.x + threadIdx.x;
    int n = tid >> 4;
    if (n >= NN) return;
    int c = (tid & 15) << 2;
    int b = batch[n];
    float4 v = *(const float4*)(XS2 + (size_t)n * FD + c);
    float* op = pool + (size_t)b * FD + c;
    atomicAdd(op + 0, v.x);
    atomicAdd(op + 1, v.y);
    atomicAdd(op + 2, v.z);
    atomicAdd(op + 3, v.w);
}

// ---------- small FC + BN (+ optional ReLU) ----------
__global__ void fc_kernel(const float* __restrict__ in, const float* __restrict__ W,
                          const float* __restrict__ b, const float* __restrict__ g,
                          const float* __restrict__ be, const float* __restrict__ rm,
                          const float* __restrict__ rv, float* __restrict__ out,
                          int rows, int K, int C, int do_relu) {
    int tid = blockIdx.x * blockDim.x + threadIdx.x;
    if (tid >= rows * C) return;
    int r = tid / C, c = tid % C;
    float acc = 0.f;
    for (int k = 0; k < K; k++) acc += in[(size_t)r * K + k] * W[(size_t)k * C + c];
    acc += b[c];
    acc = g[c] * (acc - rm[c]) * rsqrtf(rv[c] + EPS) + be[c];
    if (do_relu) acc = fmaxf(acc, 0.f);
    out[tid] = acc;
}

// =====================================================================
extern "C" void kernel_launch(void* const* d_in, const int* in_sizes, int n_in,
                              void* d_out, int out_size, void* d_ws, size_t ws_size,
                              hipStream_t stream) {
    const float* x     = (const float*)d_in[0];
    const int*   ei    = (const int*)  d_in[1];
    const int*   batch = (const int*)  d_in[2];
    const int*   sei   = (const int*)  d_in[3];   // (4,2,ES)
    const float* sea   = (const float*)d_in[4];   // (4,ES)
    const float* gW    = (const float*)d_in[5];   // (4,64,256)
    const float* gas   = (const float*)d_in[6];   // (4,4,64)
    const float* gad   = (const float*)d_in[7];
    const float* gb    = (const float*)d_in[8];   // (4,256)
    const float* mW    = (const float*)d_in[9];   // (4,256,64)
    const float* mb    = (const float*)d_in[10];
    const float* mg    = (const float*)d_in[11];
    const float* mbe   = (const float*)d_in[12];
    const float* mrm   = (const float*)d_in[13];
    const float* mrv   = (const float*)d_in[14];
    const float* f1W   = (const float*)d_in[15];
    const float* f1b   = (const float*)d_in[16];
    const float* f1g   = (const float*)d_in[17];
    const float* f1be  = (const float*)d_in[18];
    const float* f1rm  = (const float*)d_in[19];
    const float* f1rv  = (const float*)d_in[20];
    const float* f2W   = (const float*)d_in[21];
    const float* f2b   = (const float*)d_in[22];
    const float* f2g   = (const float*)d_in[23];
    const float* f2be  = (const float*)d_in[24];
    const float* f2rm  = (const float*)d_in[25];
    const float* f2rv  = (const float*)d_in[26];
    const float* f3W   = (const float*)d_in[27];
    const float* f3b   = (const float*)d_in[28];
    const float* f3g   = (const float*)d_in[29];
    const float* f3be  = (const float*)d_in[30];
    const float* f3rm  = (const float*)d_in[31];
    const float* f3rv  = (const float*)d_in[32];

    // ---- workspace layout (floats) ----
    float* ws = (float*)d_ws;
    size_t o = 0;
    float*    XS2  = ws + o; o += (size_t)NN * FD;
    float*    T0   = ws + o; o += (size_t)NN * FD;
    float*    T1   = ws + o; o += (size_t)NN * FD;
    float*    HHb  = ws + o; o += (size_t)NN * HF;
    float*    ATT  = ws + o; o += (size_t)NN * HF;
    float*    ASRC = ws + o; o += (size_t)NN * HH_;
    float*    ADST = ws + o; o += (size_t)NN * HH_;
    float*    ESC  = ws + o; o += (size_t)EA * HH_;
    unsigned* MAXV = (unsigned*)(ws + o); o += (size_t)NN * HH_;
    float*    SUMV = ws + o; o += (size_t)NN * HH_;
    float*    POOL = ws + o; o += (size_t)GG * FD;
    float*    H1   = ws + o; o += (size_t)GG * 256;
    float*    H2   = ws + o; o += (size_t)GG * FD;

    // x_stage2 accumulator starts as x
    hipMemcpyAsync(XS2, x, (size_t)NN * FD * sizeof(float),
                   hipMemcpyDeviceToDevice, stream);

    const unsigned NEG_INF_ENC = 0x007FFFFFu;  // fenc(-inf)

    for (int i = 0; i < 4; i++) {
        const int* s0 = sei + (size_t)(0 * 2 + 0) * ESN;
        const int* t0 = sei + (size_t)(0 * 2 + 1) * ESN;
        const int* si = sei + (size_t)(i * 2 + 0) * ESN;
        const int* ti = sei + (size_t)(i * 2 + 1) * ESN;
        const float* w0 = sea + (size_t)0 * ESN;
        const float* wi = sea + (size_t)i * ESN;

        // branch input -> T0
        hipMemsetAsync(T0, 0, (size_t)NN * FD * sizeof(float), stream);
        if (i == 0) {
            solo_kernel<<<(ESN * 16) / 256, 256, 0, stream>>>(s0, t0, w0, x, T0, ESN, 0);
        } else {
            hipMemsetAsync(T1, 0, (size_t)NN * FD * sizeof(float), stream);
            solo_kernel<<<(ESN * 16) / 256, 256, 0, stream>>>(si, ti, wi, x, T1, ESN, 0);
            solo_kernel<<<(ESN * 16) / 256, 256, 0, stream>>>(s0, t0, w0, T1, T0, ESN, 1);
        }

        // HH = T0 @ gW[i]   (WMMA f32)
        gemm1_wmma<<<(NN / 16) * 16 / 8, 256, 0, stream>>>(
            T0, gW + (size_t)i * FD * HF, HHb);

        // per-node scores
        scores_kernel<<<(NN * HH_ + 255) / 256, 256, 0, stream>>>(
            HHb, gas + (size_t)i * HH_ * FD, gad + (size_t)i * HH_ * FD, ASRC, ADST);

        // segment softmax over dst
        fill_u32_kernel<<<(NN * HH_ + 255) / 256, 256, 0, stream>>>(MAXV, NEG_INF_ENC, NN * HH_);
        edge_max_kernel<<<(EA * HH_ + 255) / 256, 256, 0, stream>>>(ei, ASRC, ADST, ESC, MAXV);
        hipMemsetAsync(SUMV, 0, (size_t)NN * HH_ * sizeof(float), stream);
        edge_exp_kernel<<<(EA * HH_ + 255) / 256, 256, 0, stream>>>(ei, ESC, MAXV, SUMV);

        // attention-weighted scatter
        hipMemsetAsync(ATT, 0, (size_t)NN * HF * sizeof(float), stream);
        edge_scatter_kernel<<<(EA * 64) / 256, 256, 0, stream>>>(ei, HHb, ESC, SUMV, ATT);

        // XS2 += BN( elu(ATT + gb[i]) @ mW[i] + mb[i] )   (WMMA f32, fused epilogue)
        gemm2_wmma<<<NN / 16, 128, 0, stream>>>(
            ATT, gb + (size_t)i * HF, mW + (size_t)i * HF * FD,
            mb + (size_t)i * FD, mg + (size_t)i * FD, mbe + (size_t)i * FD,
            mrm + (size_t)i * FD, mrv + (size_t)i * FD, XS2);
    }

    // global add pool over batch
    hipMemsetAsync(POOL, 0, (size_t)GG * FD * sizeof(float), stream);
    pool_kernel<<<(NN * 16) / 256, 256, 0, stream>>>(XS2, batch, POOL);

    // FC head
    fc_kernel<<<(GG * 256) / 256, 256, 0, stream>>>(POOL, f1W, f1b, f1g, f1be, f1rm, f1rv,
                                                    H1, GG, 64, 256, 1);
    fc_kernel<<<(GG * 64) / 256, 256, 0, stream>>>(H1, f2W, f2b, f2g, f2be, f2rm, f2rv,
                                                   H2, GG, 256, 64, 1);
    fc_kernel<<<(GG * 10 + 255) / 256, 256, 0, stream>>>(H2, f3W, f3b, f3g, f3be, f3rm, f3rv,
                                                         (float*)d_out, GG, 64, 10, 0);
}